// SimpleGraphSAGE_33947421508071
// MI455X (gfx1250) — compile-verified
//
#include <hip/hip_runtime.h>

#define F_IN  256
#define F_HID 128
#define EPS_BN 1e-5f

typedef float v2f __attribute__((ext_vector_type(2)));
typedef float v4f __attribute__((ext_vector_type(4)));
typedef float v8f __attribute__((ext_vector_type(8)));

// ---------------------------------------------------------------------------
// Utility: zero fill (workspace is poisoned by harness; we must clear each call)
// ---------------------------------------------------------------------------
__global__ void zero_f32(float* __restrict__ p, size_t n) {
  size_t i = (size_t)blockIdx.x * blockDim.x + threadIdx.x;
  size_t stride = (size_t)gridDim.x * blockDim.x;
  for (; i < n; i += stride) p[i] = 0.0f;
}

// ---------------------------------------------------------------------------
// Degree count: cnt[dst] += 1  (float; exact, max degree ~hundreds)
// ---------------------------------------------------------------------------
__global__ void degree_count(const int* __restrict__ dst, float* __restrict__ cnt, int E) {
  int e = blockIdx.x * blockDim.x + threadIdx.x;
  if (e < E) atomicAdd(&cnt[dst[e]], 1.0f);
}

// ---------------------------------------------------------------------------
// Segment-sum scatter: agg[dst[e]][f..f+3] += x[src[e]][f..f+3].
// One thread per (edge, float4): one global_load_b128 + 4 f32 atomics.
// Consecutive threads cover consecutive float4 groups of one edge ->
// coalesced reads and coalesced atomic bursts; agg (<=51MB) lives in the
// 192MB L2 so the f32 atomics resolve in L2's atomic units.
// ---------------------------------------------------------------------------
__global__ void scatter_add4(const float* __restrict__ x, const int* __restrict__ src,
                             const int* __restrict__ dst, float* __restrict__ agg,
                             int E, int logF4 /* log2(F/4) */, int F) {
  const size_t total = (size_t)E << logF4;
  size_t i = (size_t)blockIdx.x * blockDim.x + threadIdx.x;
  const size_t stride = (size_t)gridDim.x * blockDim.x;
  for (; i < total; i += stride) {
    const size_t e = i >> logF4;
    const int f = (int)(i & (size_t)((1 << logF4) - 1)) * 4;
    const int s = src[e];
    const int d = dst[e];
    const v4f val = *(const v4f*)(x + (size_t)s * F + f);
    float* dstp = agg + (size_t)d * F + f;
    atomicAdd(dstp + 0, val.x);
    atomicAdd(dstp + 1, val.y);
    atomicAdd(dstp + 2, val.z);
    atomicAdd(dstp + 3, val.w);
  }
}

// ---------------------------------------------------------------------------
// Fused SAGEConv combine:
//   out = BN_ReLU( (agg/cnt) @ Wl^T + xin @ Wr^T + bias )
// One wave computes a 16x16 f32 tile with V_WMMA_F32_16X16X4_F32.
// Block = 256 threads = 8 waves -> one block covers rows [m0,m0+16) x 128 cols.
//
// The 1/cnt mean scaling commutes with the GEMM (constant per output row), so
// the mean branch accumulates UNSCALED into its own accumulator and is scaled
// by rinv[row] in the epilogue. Two independent WMMA chains -> 2x ILP in the
// matrix pipe and no per-load v_pk_mul on the critical path.
//
// Fragment layouts per CDNA5 ISA (wave32):
//   A (16x4):  lane<16: M=lane, v0=K0,v1=K1 ; lane>=16: M=lane-16, v0=K2,v1=K3
//   B (4x16):  mirrored: lane holds column N=lane%16, the two K's per half-wave
//   C/D:       VGPR r: lane<16 -> (M=r, N=lane); lane>=16 -> (M=8+r, N=lane-16)
// ---------------------------------------------------------------------------
__global__ __launch_bounds__(256)
void sage_combine_wmma(const float* __restrict__ agg, const float* __restrict__ cnt,
                       const float* __restrict__ xin,
                       const float* __restrict__ Wl, const float* __restrict__ Wr,
                       const float* __restrict__ bias,
                       const float* __restrict__ bn_g, const float* __restrict__ bn_b,
                       const float* __restrict__ bn_m, const float* __restrict__ bn_v,
                       float* __restrict__ out, int K, int relu) {
  const int lane = threadIdx.x & 31;
  const int wave = threadIdx.x >> 5;
  const int l    = lane & 15;
  const int lh   = lane >> 4;        // half-wave select
  const int m0   = blockIdx.x * 16;  // output row tile
  const int n0   = wave * 16;        // output col tile (8 waves x16 = 128)
  const int arow = m0 + l;           // node row this lane streams for A
  const int brow = n0 + l;           // weight row (output feature) for B
  const int koff = 2 * lh;

  const float* aggp = agg + (size_t)arow * K + koff;
  const float* xinp = xin + (size_t)arow * K + koff;
  const float* wlp  = Wl  + (size_t)brow * K + koff;
  const float* wrp  = Wr  + (size_t)brow * K + koff;

  v8f acc_l = {};  // mean branch (unscaled)
  v8f acc_r = {};  // root branch
  for (int k0 = 0; k0 < K; k0 += 4) {
    v2f a1 = *(const v2f*)(aggp + k0);
    v2f b1 = *(const v2f*)(wlp + k0);
    acc_l = __builtin_amdgcn_wmma_f32_16x16x4_f32(false, a1, false, b1,
                                                  (short)0, acc_l, false, false);
    v2f a2 = *(const v2f*)(xinp + k0);
    v2f b2 = *(const v2f*)(wrp + k0);
    acc_r = __builtin_amdgcn_wmma_f32_16x16x4_f32(false, a2, false, b2,
                                                  (short)0, acc_r, false, false);
  }

  // Epilogue: per-row mean scale, bias + BatchNorm(eval) + ReLU, per column n.
  const int n = n0 + l;
  const float bia = bias[n];
  float scale = 1.0f, shift = 0.0f;
  if (bn_g != nullptr) {
    const float g = bn_g[n], bb = bn_b[n], m = bn_m[n], v = bn_v[n];
    scale = g * rsqrtf(v + EPS_BN);
    shift = bb - m * scale;
  }
#pragma unroll
  for (int r = 0; r < 8; ++r) {
    const int row = m0 + r + 8 * lh;
    const float rinv = 1.0f / fmaxf(cnt[row], 1.0f);
    float val = acc_l[r] * rinv + acc_r[r] + bia;
    val = val * scale + shift;
    if (relu) val = fmaxf(val, 0.0f);
    out[(size_t)row * F_HID + n] = val;
  }
}

// ---------------------------------------------------------------------------
// Decoder: gather pair = [z[s] | z[d]] (K=256), WMMA vs dw1^T, ReLU, dot with
// dw2 and LDS row-reduction (ds_add_f32) -> one scalar per pair.
// Block handles 16 pairs; 8 waves cover the 128 hidden columns.
// Two alternating accumulator chains halve the dependent-WMMA chain depth.
// ---------------------------------------------------------------------------
__global__ __launch_bounds__(256)
void decoder_wmma(const float* __restrict__ z, const int* __restrict__ eli,
                  const float* __restrict__ dw1, const float* __restrict__ db1,
                  const float* __restrict__ dw2, const float* __restrict__ db2,
                  float* __restrict__ out, int Lp) {
  __shared__ float rowsum[16];
  const int lane = threadIdx.x & 31;
  const int wave = threadIdx.x >> 5;
  const int l    = lane & 15;
  const int lh   = lane >> 4;
  const int m0   = blockIdx.x * 16;
  const int n0   = wave * 16;
  const int prow = m0 + l;
  const int brow = n0 + l;
  const int koff = 2 * lh;

  const int sidx = eli[prow];        // edge_label_index[0][row]
  const int didx = eli[Lp + prow];   // edge_label_index[1][row]
  const float* zs = z + (size_t)sidx * F_HID;
  const float* zd = z + (size_t)didx * F_HID;
  const float* wp = dw1 + (size_t)brow * (2 * F_HID) + koff;

  if (threadIdx.x < 16) rowsum[threadIdx.x] = 0.0f;
  __syncthreads();

  v8f acc0 = {};
  v8f acc1 = {};
  for (int k0 = 0; k0 < 2 * F_HID; k0 += 8) {
    const int kkA = k0 + koff;      // chunk of 4 stays within one half
    const int kkB = k0 + 4 + koff;
    v2f aA = (kkA < F_HID) ? *(const v2f*)(zs + kkA) : *(const v2f*)(zd + (kkA - F_HID));
    v2f bA = *(const v2f*)(wp + k0);
    acc0 = __builtin_amdgcn_wmma_f32_16x16x4_f32(false, aA, false, bA,
                                                 (short)0, acc0, false, false);
    v2f aB = (kkB < F_HID) ? *(const v2f*)(zs + kkB) : *(const v2f*)(zd + (kkB - F_HID));
    v2f bB = *(const v2f*)(wp + k0 + 4);
    acc1 = __builtin_amdgcn_wmma_f32_16x16x4_f32(false, aB, false, bB,
                                                 (short)0, acc1, false, false);
  }
  const v8f acc = acc0 + acc1;

  const int n = n0 + l;
  const float bia = db1[n];
  const float w2  = dw2[n];
#pragma unroll
  for (int r = 0; r < 8; ++r) {
    const float val = fmaxf(acc[r] + bia, 0.0f) * w2;
    atomicAdd(&rowsum[r + 8 * lh], val);   // ds_add_f32, row within tile
  }
  __syncthreads();
  if (threadIdx.x < 16) out[m0 + threadIdx.x] = rowsum[threadIdx.x] + db2[0];
}

// ---------------------------------------------------------------------------
// Host orchestration
// ---------------------------------------------------------------------------
extern "C" void kernel_launch(void* const* d_in, const int* in_sizes, int n_in,
                              void* d_out, int out_size, void* d_ws, size_t ws_size,
                              hipStream_t stream) {
  (void)n_in; (void)ws_size; (void)out_size;
  const float* x   = (const float*)d_in[0];
  const int*   ei  = (const int*)d_in[1];
  const int*   eli = (const int*)d_in[2];
  const float* w1l = (const float*)d_in[3];
  const float* b1l = (const float*)d_in[4];
  const float* w1r = (const float*)d_in[5];
  const float* w2l = (const float*)d_in[6];
  const float* b2l = (const float*)d_in[7];
  const float* w2r = (const float*)d_in[8];
  const float* w3l = (const float*)d_in[9];
  const float* b3l = (const float*)d_in[10];
  const float* w3r = (const float*)d_in[11];
  const float* bn1g = (const float*)d_in[12];
  const float* bn1b = (const float*)d_in[13];
  const float* bn1m = (const float*)d_in[14];
  const float* bn1v = (const float*)d_in[15];
  const float* bn2g = (const float*)d_in[16];
  const float* bn2b = (const float*)d_in[17];
  const float* bn2m = (const float*)d_in[18];
  const float* bn2v = (const float*)d_in[19];
  const float* dw1  = (const float*)d_in[20];
  const float* db1  = (const float*)d_in[21];
  const float* dw2  = (const float*)d_in[22];
  const float* db2  = (const float*)d_in[23];

  const int N = in_sizes[0] / F_IN;   // 50000
  const int E = in_sizes[1] / 2;      // 800000
  const int L = in_sizes[2] / 2;      // 200000

  const int* src = ei;
  const int* dst = ei + E;

  // Workspace layout (floats): agg[N*256] | cnt[N(pad)] | hA[N*128] | hB[N*128]
  float* ws  = (float*)d_ws;
  float* agg = ws;
  float* cnt = agg + (size_t)N * F_IN;
  float* hA  = cnt + (size_t)((N + 63) & ~63);
  float* hB  = hA + (size_t)N * F_HID;

  const int T = 256;
  const int gemm_grid = N / 16;            // 3125 (N divisible by 16)
  auto sgrid = [](size_t total) {
    size_t b = (total + 255) / 256;
    return (int)(b > 262144 ? 262144 : b); // grid-stride handles the rest
  };

  // ---- Layer 1: agg/cnt over x (K=256), combine + BN1 + ReLU -> hA
  zero_f32<<<sgrid((size_t)N * F_IN), T, 0, stream>>>(agg, (size_t)N * F_IN);
  zero_f32<<<sgrid(N), T, 0, stream>>>(cnt, (size_t)N);
  degree_count<<<(E + T - 1) / T, T, 0, stream>>>(dst, cnt, E);
  scatter_add4<<<sgrid((size_t)E << 6), T, 0, stream>>>(x, src, dst, agg, E, 6, F_IN);
  sage_combine_wmma<<<gemm_grid, T, 0, stream>>>(agg, cnt, x, w1l, w1r, b1l,
                                                 bn1g, bn1b, bn1m, bn1v,
                                                 hA, F_IN, 1);

  // ---- Layer 2: agg over hA (K=128), combine + BN2 + ReLU -> hB
  zero_f32<<<sgrid((size_t)N * F_HID), T, 0, stream>>>(agg, (size_t)N * F_HID);
  scatter_add4<<<sgrid((size_t)E << 5), T, 0, stream>>>(hA, src, dst, agg, E, 5, F_HID);
  sage_combine_wmma<<<gemm_grid, T, 0, stream>>>(agg, cnt, hA, w2l, w2r, b2l,
                                                 bn2g, bn2b, bn2m, bn2v,
                                                 hB, F_HID, 1);

  // ---- Layer 3: agg over hB (K=128), combine (no BN / no ReLU) -> z in hA
  zero_f32<<<sgrid((size_t)N * F_HID), T, 0, stream>>>(agg, (size_t)N * F_HID);
  scatter_add4<<<sgrid((size_t)E << 5), T, 0, stream>>>(hB, src, dst, agg, E, 5, F_HID);
  sage_combine_wmma<<<gemm_grid, T, 0, stream>>>(agg, cnt, hB, w3l, w3r, b3l,
                                                 nullptr, nullptr, nullptr, nullptr,
                                                 hA, F_HID, 0);

  // ---- Decoder: 200000 pairs, 16 per block
  decoder_wmma<<<L / 16, T, 0, stream>>>(hA, eli, dw1, db1, dw2, db2,
                                         (float*)d_out, L);
}